// WeaveLayer_28982439313937
// MI455X (gfx1250) — compile-verified
//
#include <hip/hip_runtime.h>

// WeaveLayer forward for MI455X (gfx1250), wave32 + V_WMMA_F32_16X16X4_F32.
//
// Key algebraic optimization: the gathered GEMM
//   AP = relu(a_ij @ W_AP + b) + relu(a_ji @ W_AP + b)
// factorizes via W_AP = [W_top; W_bot] into per-atom projections
//   X1 = atom @ W_top, X2 = atom @ W_bot            (1.5 GFLOP, done once)
//   AP[p] = relu(X1[i]+X2[j]+b) + relu(X1[j]+X2[i]+b)
// turning 48 GFLOP of gather-GEMM into gathers from a 40MB L2-resident table.

typedef __attribute__((ext_vector_type(2))) float v2f;
typedef __attribute__((ext_vector_type(8))) float v8f;

#define DEVINL __device__ __forceinline__

constexpr int N_ATOMS = 100000;
constexpr int N_PAIRS = 1600000;
constexpr int F_A = 75;   // atom feats (pad K to 76)
constexpr int F_P = 14;   // pair feats (pad K to 16)
constexpr int H   = 50;   // hidden / output width (pad N to 64)

DEVINL float relu(float x) { return x > 0.f ? x : 0.f; }

// One 16x16x4 f32 WMMA step. Per ISA layout:
//  A frag: lanes 0-15 hold A[M=l][k..k+1], lanes 16-31 hold A[l][k+2..k+3]
//  B frag: lanes 0-15 hold B[k..k+1][N=l], lanes 16-31 hold B[k+2..k+3][l]
//  C/D:    vgpr r, lane l, half h -> D[M = r + 8h][N = l]
DEVINL v8f wmma4(v2f a, v2f b, v8f c) {
  return __builtin_amdgcn_wmma_f32_16x16x4_f32(
      /*neg_a=*/false, a, /*neg_b=*/false, b,
      /*c_mod=*/(short)0, c, /*reuse_a=*/false, /*reuse_b=*/false);
}

// Accumulate a full 16x16 tile: lA is LDS A-tile [16][ldA], lB is LDS B [K][64],
// nb selects the 16-col N-tile. K must be a multiple of 4.
DEVINL v8f wmma_tile(const float* lA, int ldA, const float* lB, int nb,
                     int K, int l, int half, v8f c) {
  for (int k = 0; k < K; k += 4) {
    const int ka = k + half * 2;
    v2f a, b;
    a.x = lA[l * ldA + ka];
    a.y = lA[l * ldA + ka + 1];
    b.x = lB[ka * 64 + nb + l];
    b.y = lB[(ka + 1) * 64 + nb + l];
    c = wmma4(a, b, c);
  }
  return c;
}

__global__ void zero_f32(float* __restrict__ p, int n) {
  int i = blockIdx.x * blockDim.x + threadIdx.x;
  if (i < n) p[i] = 0.f;
}

// ---------------------------------------------------------------------------
// Kernel 2: X1 = atom @ W_AP[0:75], X2 = atom @ W_AP[75:150]  (no bias/relu)
// 128 threads = 4 waves; 64 atoms per block, each wave one 16-row M-tile.
// ---------------------------------------------------------------------------
__global__ __launch_bounds__(128) void atom_proj_kernel(
    const float* __restrict__ atomF, const float* __restrict__ W_AP,
    float* __restrict__ X1, float* __restrict__ X2) {
  __shared__ float sA[64 * 76];    // atom tile, K padded 75->76
  __shared__ float sW1[76 * 64];   // W_AP top half, N padded 50->64
  __shared__ float sW2[76 * 64];   // W_AP bottom half
  const int tid = threadIdx.x;
  const int base = blockIdx.x * 64;

  for (int e = tid; e < 64 * 76; e += 128) {
    int r = e / 76, c = e % 76;
    int a = base + r;
    sA[e] = (a < N_ATOMS && c < F_A) ? atomF[a * F_A + c] : 0.f;
  }
  for (int e = tid; e < 76 * 64; e += 128) {
    int r = e >> 6, c = e & 63;
    float w1 = 0.f, w2 = 0.f;
    if (r < F_A && c < H) {
      w1 = W_AP[r * H + c];
      w2 = W_AP[(F_A + r) * H + c];
    }
    sW1[e] = w1;
    sW2[e] = w2;
  }
  __syncthreads();

  const int lane = tid & 31, wave = tid >> 5;
  const int l = lane & 15, half = lane >> 4;
  const float* lA = &sA[wave * 16 * 76];

  for (int nt = 0; nt < 4; ++nt) {
    const int nb = nt * 16;
    v8f c1 = {}, c2 = {};
    for (int k = 0; k < 76; k += 4) {
      const int ka = k + half * 2;
      v2f a;  a.x  = lA[l * 76 + ka];        a.y  = lA[l * 76 + ka + 1];
      v2f b1; b1.x = sW1[ka * 64 + nb + l];  b1.y = sW1[(ka + 1) * 64 + nb + l];
      v2f b2; b2.x = sW2[ka * 64 + nb + l];  b2.y = sW2[(ka + 1) * 64 + nb + l];
      c1 = wmma4(a, b1, c1);
      c2 = wmma4(a, b2, c2);
    }
    const int col = nb + l;
    if (col < H) {
#pragma unroll
      for (int r = 0; r < 8; ++r) {
        const int atom = base + wave * 16 + r + 8 * half;
        if (atom < N_ATOMS) {
          X1[atom * H + col] = c1[r];
          X2[atom * H + col] = c2[r];
        }
      }
    }
  }
}

// ---------------------------------------------------------------------------
// Kernel 3: per 64-pair tile:
//   PP = relu(pair @ W_PP + b_PP)                      (WMMA, K=16)
//   PA = relu(pair @ W_PA + b_PA) -> atomicAdd segsum  (WMMA, K=16)
//   AP = relu(X1[i]+X2[j]+b_AP) + relu(X1[j]+X2[i]+b_AP)   (gather, L2-hot)
//   P  = relu([AP|PP] @ W_P + b_P)                     (WMMA, K=100)
// ---------------------------------------------------------------------------
__global__ __launch_bounds__(128) void pair_kernel(
    const float* __restrict__ pairF, const int* __restrict__ pairSplit,
    const int* __restrict__ a2p,
    const float* __restrict__ W_PA, const float* __restrict__ b_PA,
    const float* __restrict__ W_PP, const float* __restrict__ b_PP,
    const float* __restrict__ W_P,  const float* __restrict__ b_P,
    const float* __restrict__ b_AP,
    const float* __restrict__ X1, const float* __restrict__ X2,
    float* __restrict__ PA, float* __restrict__ outP) {
  __shared__ float sPair[64 * 16];    // pair tile, K padded 14->16
  __shared__ float sWpa[16 * 64];
  __shared__ float sWpp[16 * 64];
  __shared__ float sWp[100 * 64];     // [W_P top; W_P bottom], K=100
  __shared__ float sAt[64 * 100];     // concat A-tile [AP(0..49) | PP(50..99)]
  __shared__ float sBpa[64], sBpp[64], sBp[64], sBap[64];

  const int tid = threadIdx.x;
  const int base = blockIdx.x * 64;   // N_PAIRS % 64 == 0

  for (int e = tid; e < 64 * 16; e += 128) {
    int r = e >> 4, c = e & 15;
    sPair[e] = (c < F_P) ? pairF[(base + r) * F_P + c] : 0.f;
  }
  for (int e = tid; e < 16 * 64; e += 128) {
    int r = e >> 6, c = e & 63;
    float wpa = 0.f, wpp = 0.f;
    if (r < F_P && c < H) { wpa = W_PA[r * H + c]; wpp = W_PP[r * H + c]; }
    sWpa[e] = wpa;
    sWpp[e] = wpp;
  }
  for (int e = tid; e < 100 * 64; e += 128) {
    int r = e >> 6, c = e & 63;
    sWp[e] = (c < H) ? W_P[r * H + c] : 0.f;   // rows 0..49 -> AP, 50..99 -> PP
  }
  if (tid < 64) {
    sBpa[tid] = (tid < H) ? b_PA[tid] : 0.f;
    sBpp[tid] = (tid < H) ? b_PP[tid] : 0.f;
    sBp[tid]  = (tid < H) ? b_P[tid]  : 0.f;
    sBap[tid] = (tid < H) ? b_AP[tid] : 0.f;
  }
  __syncthreads();

  const int lane = tid & 31, wave = tid >> 5;
  const int l = lane & 15, half = lane >> 4;
  const float* lA = &sPair[wave * 16 * 16];

  // PP and PA (K = 16)
  for (int nt = 0; nt < 4; ++nt) {
    const int nb = nt * 16;
    v8f cp = {}, ca = {};
    for (int k = 0; k < 16; k += 4) {
      const int ka = k + half * 2;
      v2f a;  a.x  = lA[l * 16 + ka];         a.y  = lA[l * 16 + ka + 1];
      v2f bp; bp.x = sWpp[ka * 64 + nb + l];  bp.y = sWpp[(ka + 1) * 64 + nb + l];
      v2f ba; ba.x = sWpa[ka * 64 + nb + l];  ba.y = sWpa[(ka + 1) * 64 + nb + l];
      cp = wmma4(a, bp, cp);
      ca = wmma4(a, ba, ca);
    }
    const int col = nb + l;
    if (col < H) {
#pragma unroll
      for (int r = 0; r < 8; ++r) {
        const int row = wave * 16 + r + 8 * half;
        sAt[row * 100 + 50 + col] = relu(cp[r] + sBpp[col]);   // PP into concat
        const int seg = pairSplit[base + row];
        atomicAdd(&PA[seg * H + col], relu(ca[r] + sBpa[col])); // PA segment sum
      }
    }
  }

  // AP via factorized gathers (X1/X2 live in L2)
  for (int e = tid; e < 64 * H; e += 128) {
    const int p = e / H, c = e % H;
    const int gi = a2p[2 * (base + p)];
    const int gj = a2p[2 * (base + p) + 1];
    const float bap = sBap[c];
    const float v1 = relu(X1[gi * H + c] + X2[gj * H + c] + bap);
    const float v2 = relu(X1[gj * H + c] + X2[gi * H + c] + bap);
    sAt[p * 100 + c] = v1 + v2;
  }
  __syncthreads();

  // P = relu([AP|PP] @ W_P + b_P), K = 100
  const float* lA2 = &sAt[wave * 16 * 100];
  for (int nt = 0; nt < 4; ++nt) {
    const int nb = nt * 16;
    v8f c = wmma_tile(lA2, 100, sWp, nb, 100, l, half, v8f{});
    const int col = nb + l;
    if (col < H) {
#pragma unroll
      for (int r = 0; r < 8; ++r) {
        const int row = wave * 16 + r + 8 * half;
        outP[(size_t)(base + row) * H + col] = relu(c[r] + sBp[col]);
      }
    }
  }
}

// ---------------------------------------------------------------------------
// Kernel 4: AA = relu(atom @ W_AA + b_AA) recomputed on the fly, then
//           A = relu([AA|PA] @ W_A + b_A).  Phased LDS reuse (<64KB static).
// ---------------------------------------------------------------------------
__global__ __launch_bounds__(128) void atom_out_kernel(
    const float* __restrict__ atomF, const float* __restrict__ PA,
    const float* __restrict__ W_AA, const float* __restrict__ b_AA,
    const float* __restrict__ W_A,  const float* __restrict__ b_A,
    float* __restrict__ outA) {
  __shared__ float sBig[9728];        // phase1: atoms[64*76] + W_AA[76*64]; phase2: W_A[100*64]
  __shared__ float sAt[64 * 100];     // concat [AA(0..49) | PA(50..99)]
  __shared__ float sBaa[64], sBa[64];
  float* sAtom = sBig;                // 4864 floats
  float* sWaa  = sBig + 64 * 76;      // 4864 floats
  float* sWa   = sBig;                // 6400 floats (phase 2, reuses region)

  const int tid = threadIdx.x;
  const int base = blockIdx.x * 64;

  for (int e = tid; e < 64 * 76; e += 128) {
    int r = e / 76, c = e % 76;
    int a = base + r;
    sAtom[e] = (a < N_ATOMS && c < F_A) ? atomF[a * F_A + c] : 0.f;
  }
  for (int e = tid; e < 76 * 64; e += 128) {
    int r = e >> 6, c = e & 63;
    sWaa[e] = (r < F_A && c < H) ? W_AA[r * H + c] : 0.f;
  }
  if (tid < 64) {
    sBaa[tid] = (tid < H) ? b_AA[tid] : 0.f;
    sBa[tid]  = (tid < H) ? b_A[tid]  : 0.f;
  }
  __syncthreads();

  const int lane = tid & 31, wave = tid >> 5;
  const int l = lane & 15, half = lane >> 4;
  const float* lA = &sAtom[wave * 16 * 76];

  // AA (K = 76) -> concat cols 0..49
  for (int nt = 0; nt < 4; ++nt) {
    const int nb = nt * 16;
    v8f c = wmma_tile(lA, 76, sWaa, nb, 76, l, half, v8f{});
    const int col = nb + l;
    if (col < H) {
#pragma unroll
      for (int r = 0; r < 8; ++r) {
        const int row = wave * 16 + r + 8 * half;
        sAt[row * 100 + col] = relu(c[r] + sBaa[col]);
      }
    }
  }
  // PA -> concat cols 50..99 (segment sums already relu'd pre-sum per reference)
  for (int e = tid; e < 64 * H; e += 128) {
    const int p = e / H, c = e % H;
    const int a = base + p;
    sAt[p * 100 + 50 + c] = (a < N_ATOMS) ? PA[a * H + c] : 0.f;
  }
  __syncthreads();

  // Phase 2: reuse sBig for W_A
  for (int e = tid; e < 100 * 64; e += 128) {
    int r = e >> 6, c = e & 63;
    sWa[e] = (c < H) ? W_A[r * H + c] : 0.f;
  }
  __syncthreads();

  const float* lA2 = &sAt[wave * 16 * 100];
  for (int nt = 0; nt < 4; ++nt) {
    const int nb = nt * 16;
    v8f c = wmma_tile(lA2, 100, sWa, nb, 100, l, half, v8f{});
    const int col = nb + l;
    if (col < H) {
#pragma unroll
      for (int r = 0; r < 8; ++r) {
        const int atom = base + wave * 16 + r + 8 * half;
        if (atom < N_ATOMS)
          outA[(size_t)atom * H + col] = relu(c[r] + sBa[col]);
      }
    }
  }
}

extern "C" void kernel_launch(void* const* d_in, const int* in_sizes, int n_in,
                              void* d_out, int out_size, void* d_ws, size_t ws_size,
                              hipStream_t stream) {
  const float* atomF     = (const float*)d_in[0];
  const float* pairF     = (const float*)d_in[1];
  const int*   pairSplit = (const int*)d_in[2];
  const int*   a2p       = (const int*)d_in[3];
  const float* W_AA = (const float*)d_in[4];
  const float* b_AA = (const float*)d_in[5];
  const float* W_PA = (const float*)d_in[6];
  const float* b_PA = (const float*)d_in[7];
  const float* W_A  = (const float*)d_in[8];
  const float* b_A  = (const float*)d_in[9];
  const float* W_AP = (const float*)d_in[10];
  const float* b_AP = (const float*)d_in[11];
  const float* W_PP = (const float*)d_in[12];
  const float* b_PP = (const float*)d_in[13];
  const float* W_P  = (const float*)d_in[14];
  const float* b_P  = (const float*)d_in[15];

  float* outA = (float*)d_out;                    // [N_ATOMS * H]
  float* outP = outA + (size_t)N_ATOMS * H;       // [N_PAIRS * H]

  float* X1 = (float*)d_ws;                       // [N_ATOMS * H]
  float* X2 = X1 + (size_t)N_ATOMS * H;           // [N_ATOMS * H]
  float* PA = X2 + (size_t)N_ATOMS * H;           // [N_ATOMS * H] segment accum

  const int nPA = N_ATOMS * H;
  zero_f32<<<(nPA + 255) / 256, 256, 0, stream>>>(PA, nPA);
  atom_proj_kernel<<<(N_ATOMS + 63) / 64, 128, 0, stream>>>(atomF, W_AP, X1, X2);
  pair_kernel<<<N_PAIRS / 64, 128, 0, stream>>>(
      pairF, pairSplit, a2p, W_PA, b_PA, W_PP, b_PP, W_P, b_P, b_AP,
      X1, X2, PA, outP);
  atom_out_kernel<<<(N_ATOMS + 63) / 64, 128, 0, stream>>>(
      atomF, PA, W_AA, b_AA, W_A, b_A, outA);
}